// AutoregressiveGRUWithAttention_25434796327695
// MI455X (gfx1250) — compile-verified
//
#include <hip/hip_runtime.h>
#include <hip/hip_bf16.h>
#include <math.h>

typedef __attribute__((ext_vector_type(2))) float v2f;
typedef __attribute__((ext_vector_type(8))) float v8f;

#define B_TOT    1024
#define L_ENC    64
#define T_DEC    128
#define IN_F     13
#define H_DIM    64
#define OUT_F    13
#define G3       192      // 3*H
#define TILE_B   16
#define NTHREADS 128
#define NWAVES   4

// Row strides (floats). Chosen for conflict-free wave32 WMMA fragment loads:
//  - A-operand row stride 68 (= 4 mod 64): every lane's b64 A-load hits a unique bank pair
//  - B pair-row strides = 32 mod 64: lanes 0-15 and 16-31 land in disjoint bank halves
#define LDW 196   // 192-column gate buffers (not WMMA operands)
#define LDH 68    // 64-column A-operand / state buffers
#define LDX 20    // 13->16 padded input/output tiles (A operand)
#define PSW 416   // pair-row stride for 192-wide B (2*192=384, padded)
#define PSA 160   // pair-row stride for  64-wide B (2*64=128, padded)
#define PSF 32    // pair-row stride for  16-wide B (2*16=32)

// ---- LDS layout (float offsets) ----
#define OF_WIHP 0                        // [8][PSW]   Wih^T pair-interleaved (k 0..15, pads 0)
#define OF_WHHP (OF_WIHP + 8*PSW)        // [32][PSW]  Whh^T pair-interleaved
#define OF_WAP  (OF_WHHP + 32*PSW)       // [32][PSA]  Wa^T  pair-interleaved
#define OF_WFP  (OF_WAP  + 32*PSA)       // [32][PSF]  Wf^T  pair-interleaved (cols 13..15 zero)
#define OF_BIH  (OF_WFP  + 32*PSF)       // [192]
#define OF_BHH  (OF_BIH  + G3)           // [192]
#define OF_BA   (OF_BHH  + G3)           // [64]
#define OF_BF   (OF_BA   + H_DIM)        // [16] (13 valid, rest 0)
#define OF_XP   (OF_BF   + 16)           // [16][LDX]  x_t tile, cols 13..15 zero
#define OF_YP   (OF_XP   + 16*LDX)       // [16][LDX]  y_{t-1} tile (decoder input)
#define OF_H    (OF_YP   + 16*LDX)       // [16][LDH]  hidden state
#define OF_O    (OF_H    + 16*LDH)       // [16][LDH]  attention output o_t
#define OF_GI   (OF_O    + 16*LDH)       // [16][LDW]  input gates
#define OF_GH   (OF_GI   + 16*LDW)       // [16][LDW]  hidden gates
#define OF_PROJ (OF_GH   + 16*LDW)       // [16][LDH]  o_t @ Wa^T + ba
#define OF_MX   (OF_PROJ + 16*LDH)       // [16][LDH]  online softmax running max
#define OF_DEN  (OF_MX   + 16*LDH)       // [16][LDH]  online softmax denominator
#define OF_NUM  (OF_DEN  + 16*LDH)       // [16][LDH]  online softmax numerator
#define OF_LEN  (OF_NUM  + 16*LDH)       // [16] int lengths
#define SMEM_FLOATS (OF_LEN + 16)

// D = A[16xK] * B[KxN] + bias; B pair-interleaved: BP[(k>>1)*ps + 2n + (k&1)] = B[k][n].
// f32 WMMA fragment layout per CDNA5 ISA 7.12.2:
//  A 16x4: lane m=lane&15, half hh=lane>>4 holds A[m][kb+2hh], A[m][kb+2hh+1] (contig -> b64)
//  B 4x16: lane n holds B[kb+2hh][n], B[kb+2hh+1][n] (contig in pair layout -> b64)
//  C/D 16x16: vgpr v, lane -> D[v+8*hh][lane&15]
template<int K, int NTILES>
__device__ __forceinline__ void wmma_gemm_f32(
    const float* __restrict__ A, const int lda,
    const float* __restrict__ BP, const int psb,
    float* __restrict__ Out, const int ldo,
    const float* __restrict__ bias,
    const int lane, const int wave)
{
    const int m  = lane & 15;
    const int hh = lane >> 4;
#pragma unroll
    for (int i = 0; i < (NTILES + NWAVES - 1) / NWAVES; ++i) {
        const int nt = wave + i * NWAVES;
        if (nt < NTILES) {                      // wave-uniform; folds when NTILES%4==0
            const int nc  = nt * 16 + m;
            const int nc2 = 2 * nc;
            v8f acc = {};
#pragma unroll
            for (int kb = 0; kb < K; kb += 4) {
                v2f av = *(const v2f*)(A + m * lda + kb + 2 * hh);
                v2f bv = *(const v2f*)(BP + (kb / 2 + hh) * psb + nc2);
                acc = __builtin_amdgcn_wmma_f32_16x16x4_f32(
                    false, av, false, bv, (short)0, acc, false, false);
            }
            const float bb = bias[nc];
#pragma unroll
            for (int v = 0; v < 8; ++v)
                Out[(v + 8 * hh) * ldo + nc] = acc[v] + bb;
        }
    }
}

__device__ __forceinline__ float sigmoidf_fast(float v) {
    return 1.0f / (1.0f + __expf(-v));
}

__global__ __launch_bounds__(NTHREADS)
void argru_attn_persistent(const float* __restrict__ x,
                           const int*   __restrict__ lengths,
                           const float* __restrict__ Wih,
                           const float* __restrict__ Whh,
                           const float* __restrict__ bih,
                           const float* __restrict__ bhh,
                           const float* __restrict__ Wf,
                           const float* __restrict__ bf,
                           const float* __restrict__ Wa,
                           const float* __restrict__ ba,
                           float* __restrict__ out)
{
    extern __shared__ float sm[];
    const int tid  = threadIdx.x;
    const int lane = tid & 31;
    const int wave = tid >> 5;
    __builtin_assume((unsigned)wave < NWAVES);
    const int bBase = blockIdx.x * TILE_B;

    float* sWihP = sm + OF_WIHP;
    float* sWhhP = sm + OF_WHHP;
    float* sWaP  = sm + OF_WAP;
    float* sWfP  = sm + OF_WFP;
    float* sBih  = sm + OF_BIH;
    float* sBhh  = sm + OF_BHH;
    float* sBa   = sm + OF_BA;
    float* sBf   = sm + OF_BF;
    float* sXP   = sm + OF_XP;
    float* sYP   = sm + OF_YP;
    float* sH    = sm + OF_H;
    float* sO    = sm + OF_O;
    float* sGI   = sm + OF_GI;
    float* sGH   = sm + OF_GH;
    float* sProj = sm + OF_PROJ;
    float* sMx   = sm + OF_MX;
    float* sDen  = sm + OF_DEN;
    float* sNum  = sm + OF_NUM;
    int*   sLen  = (int*)(sm + OF_LEN);

    // ---- preamble: zero LDS (provides every pad), then stage weights ----
    for (int i = tid; i < SMEM_FLOATS; i += NTHREADS) sm[i] = 0.0f;
    __syncthreads();
    // pair-interleaved B stage: src row-major W[n][k] -> BP[(k>>1)*ps + 2n + (k&1)]
    for (int idx = tid; idx < G3 * IN_F; idx += NTHREADS) {
        int n = idx / IN_F, k = idx % IN_F;
        sWihP[(k >> 1) * PSW + 2 * n + (k & 1)] = Wih[idx];
    }
    for (int idx = tid; idx < G3 * H_DIM; idx += NTHREADS) {
        int n = idx / H_DIM, k = idx % H_DIM;
        sWhhP[(k >> 1) * PSW + 2 * n + (k & 1)] = Whh[idx];
    }
    for (int idx = tid; idx < H_DIM * H_DIM; idx += NTHREADS) {
        int n = idx / H_DIM, k = idx % H_DIM;
        sWaP[(k >> 1) * PSA + 2 * n + (k & 1)] = Wa[idx];
    }
    for (int idx = tid; idx < OUT_F * H_DIM; idx += NTHREADS) {
        int n = idx / H_DIM, k = idx % H_DIM;
        sWfP[(k >> 1) * PSF + 2 * n + (k & 1)] = Wf[idx];
    }
    for (int i = tid; i < G3; i += NTHREADS) { sBih[i] = bih[i]; sBhh[i] = bhh[i]; }
    for (int i = tid; i < H_DIM; i += NTHREADS) sBa[i] = ba[i];
    for (int i = tid; i < OUT_F; i += NTHREADS) sBf[i] = bf[i];
    for (int i = tid; i < TILE_B; i += NTHREADS) sLen[i] = lengths[bBase + i];
    for (int i = tid; i < TILE_B * LDH; i += NTHREADS) sMx[i] = -__builtin_inff();
    __syncthreads();

    // =================== encoder: 64 masked GRU steps ===================
    for (int t = 0; t < L_ENC; ++t) {
        for (int idx = tid; idx < TILE_B * IN_F; idx += NTHREADS) {
            int b = idx / IN_F, i = idx % IN_F;
            sXP[b * LDX + i] = x[(size_t)(bBase + b) * (L_ENC * IN_F) + t * IN_F + i];
        }
        __syncthreads();
        wmma_gemm_f32<16, 12>(sXP, LDX, sWihP, PSW, sGI, LDW, sBih, lane, wave);
        wmma_gemm_f32<64, 12>(sH,  LDH, sWhhP, PSW, sGH, LDW, sBhh, lane, wave);
        __syncthreads();
        const bool last = (t == L_ENC - 1);
        for (int e = tid; e < TILE_B * H_DIM; e += NTHREADS) {
            int b = e >> 6, k = e & 63;
            float r  = sigmoidf_fast(sGI[b*LDW + k]      + sGH[b*LDW + k]);
            float z  = sigmoidf_fast(sGI[b*LDW + 64 + k] + sGH[b*LDW + 64 + k]);
            float n  = tanhf(sGI[b*LDW + 128 + k] + r * sGH[b*LDW + 128 + k]);
            float hp = sH[b*LDH + k];
            float hn = (1.0f - z) * n + z * hp;
            bool valid = t < sLen[b];
            sH[b*LDH + k] = valid ? hn : hp;
            if (last) sO[b*LDH + k] = valid ? hn : 0.0f;   // outs[-1]
        }
        __syncthreads();
    }
    // nin = outs[-1] @ Wf^T + bf -> initial decoder input (cols 13..15 exact 0)
    wmma_gemm_f32<64, 1>(sO, LDH, sWfP, PSF, sYP, LDX, sBf, lane, wave);
    __syncthreads();

    // =================== decoder: 128 steps, online-softmax attention ===================
    for (int t = 0; t < T_DEC; ++t) {
        wmma_gemm_f32<16, 12>(sYP, LDX, sWihP, PSW, sGI, LDW, sBih, lane, wave);
        wmma_gemm_f32<64, 12>(sH,  LDH, sWhhP, PSW, sGH, LDW, sBhh, lane, wave);
        __syncthreads();
        for (int e = tid; e < TILE_B * H_DIM; e += NTHREADS) {
            int b = e >> 6, k = e & 63;
            float r  = sigmoidf_fast(sGI[b*LDW + k]      + sGH[b*LDW + k]);
            float z  = sigmoidf_fast(sGI[b*LDW + 64 + k] + sGH[b*LDW + 64 + k]);
            float n  = tanhf(sGI[b*LDW + 128 + k] + r * sGH[b*LDW + 128 + k]);
            float hp = sH[b*LDH + k];
            float hn = (1.0f - z) * n + z * hp;
            sH[b*LDH + k] = hn;
            // attention over entries 0..t-1 via running numerator/denominator
            float den = sDen[b*LDH + k];
            float att = (den > 0.0f) ? (sNum[b*LDH + k] / den) : 0.0f;
            sO[b*LDH + k] = hn + att;
        }
        __syncthreads();
        // proj_t = o_t @ Wa^T + ba (entry for future steps); y_t = o_t @ Wf^T + bf
        wmma_gemm_f32<64, 4>(sO, LDH, sWaP, PSA, sProj, LDH, sBa, lane, wave);
        wmma_gemm_f32<64, 1>(sO, LDH, sWfP, PSF, sYP,   LDX, sBf, lane, wave);
        __syncthreads();
        // emit y_t, fold entry t into online softmax state (used from step t+1 on)
        for (int idx = tid; idx < TILE_B * OUT_F; idx += NTHREADS) {
            int b = idx / OUT_F, o = idx % OUT_F;
            out[(size_t)(bBase + b) * (T_DEC * OUT_F) + (size_t)t * OUT_F + o] =
                sYP[b * LDX + o];
        }
        for (int e = tid; e < TILE_B * H_DIM; e += NTHREADS) {
            int b = e >> 6, g = e & 63;
            float p  = sProj[b*LDH + g];
            float mo = sMx[b*LDH + g];
            float mn = fmaxf(mo, p);
            float sc = __expf(mo - mn);      // exp(-inf)=0 on first entry
            float w  = __expf(p - mn);
            sDen[b*LDH + g] = sDen[b*LDH + g] * sc + w;
            sNum[b*LDH + g] = sNum[b*LDH + g] * sc + w * sO[b*LDH + g];
            sMx[b*LDH + g]  = mn;
        }
        __syncthreads();
    }
}

extern "C" void kernel_launch(void* const* d_in, const int* in_sizes, int n_in,
                              void* d_out, int out_size, void* d_ws, size_t ws_size,
                              hipStream_t stream) {
    (void)in_sizes; (void)n_in; (void)out_size; (void)d_ws; (void)ws_size;
    const float* x       = (const float*)d_in[0];
    const int*   lengths = (const int*)d_in[1];
    // d_in[2] = output_length scalar (T_DEC fixed at compile time)
    const float* Wih = (const float*)d_in[3];
    const float* Whh = (const float*)d_in[4];
    const float* bih = (const float*)d_in[5];
    const float* bhh = (const float*)d_in[6];
    const float* Wf  = (const float*)d_in[7];
    const float* bf  = (const float*)d_in[8];
    const float* Wa  = (const float*)d_in[9];
    const float* ba  = (const float*)d_in[10];
    float* out = (float*)d_out;

    const size_t smemBytes = (size_t)SMEM_FLOATS * sizeof(float);
    dim3 grid(B_TOT / TILE_B);   // 64 independent batch-tile workers
    dim3 block(NTHREADS);
    argru_attn_persistent<<<grid, block, smemBytes, stream>>>(
        x, lengths, Wih, Whh, bih, bhh, Wf, bf, Wa, ba, out);
}